// GATLayer_30339648979520
// MI455X (gfx1250) — compile-verified
//
#include <hip/hip_runtime.h>
#include <hip/hip_bf16.h>

typedef __attribute__((ext_vector_type(2))) float v2f;
typedef __attribute__((ext_vector_type(8))) float v8f;

#define SLOPE 0.2f

// ---------------------------------------------------------------------------
// Kernel 1: xp[N,128] = x[N,128] @ W[128,128] via V_WMMA_F32_16X16X4_F32.
// One wave computes one 16x16 tile of xp; 8 waves/block = 16x128 row strip.
// A-frag (16x4 f32): lane m=l&15, VGPR0/1 hold K = 2*(l>>4)+0/1.
// B-frag (4x16 f32): lane n=l&15, VGPR0/1 hold K = 2*(l>>4)+0/1.
// D (16x16 f32, 8 VGPRs): VGPR v -> row M = v + 8*(l>>4), col N = l&15.
// ---------------------------------------------------------------------------
__global__ void gat_gemm_wmma(const float* __restrict__ x,
                              const float* __restrict__ W,
                              float* __restrict__ xp, int n) {
    const int lane = threadIdx.x & 31;
    const int wave = threadIdx.x >> 5;          // 0..7 -> column tile
    const int row_base = blockIdx.x * 16;
    if (row_base >= n) return;                   // uniform per block
    const int col_base = wave * 16;
    const int m  = lane & 15;
    const int kk = (lane >> 4) * 2;

    int arow = row_base + m;
    if (arow > n - 1) arow = n - 1;              // clamp (keeps EXEC all-ones)
    const float* __restrict__ xrow = x + (size_t)arow * 128;

    v8f acc = {};
    #pragma unroll
    for (int k0 = 0; k0 < 128; k0 += 4) {
        v2f a, b;
        a.x = xrow[k0 + kk];
        a.y = xrow[k0 + kk + 1];
        b.x = W[(size_t)(k0 + kk)     * 128 + col_base + m];
        b.y = W[(size_t)(k0 + kk + 1) * 128 + col_base + m];
        acc = __builtin_amdgcn_wmma_f32_16x16x4_f32(
                  false, a, false, b, (short)0, acc, false, false);
    }

    const int half = lane >> 4;
    #pragma unroll
    for (int v = 0; v < 8; ++v) {
        int row = row_base + v + 8 * half;
        if (row < n)
            xp[(size_t)row * 128 + col_base + m] = acc[v];
    }
}

// ---------------------------------------------------------------------------
// Kernel 2: per-(node,head) attention logits
// a_src[n,h] = sum_c xp[n,h,c]*att_src[h,c]   (same for dst)
// ---------------------------------------------------------------------------
__global__ void gat_logits(const float* __restrict__ xp,
                           const float* __restrict__ att_src,
                           const float* __restrict__ att_dst,
                           float* __restrict__ a_src,
                           float* __restrict__ a_dst, int n) {
    int t = blockIdx.x * blockDim.x + threadIdx.x;
    if (t >= n * 4) return;
    int node = t >> 2, h = t & 3;
    const float* __restrict__ v  = xp + (size_t)node * 128 + h * 32;
    const float* __restrict__ as = att_src + h * 32;
    const float* __restrict__ ad = att_dst + h * 32;
    float s = 0.f, d = 0.f;
    #pragma unroll
    for (int c = 0; c < 32; ++c) {
        float xv = v[c];
        s = fmaf(xv, as[c], s);
        d = fmaf(xv, ad[c], d);
    }
    a_src[t] = s;
    a_dst[t] = d;
}

// ---------------------------------------------------------------------------
// Kernel 3: init accumulators (agg=0, emax=-inf, denom=0)
// ---------------------------------------------------------------------------
__global__ void gat_init(float* __restrict__ agg, float* __restrict__ emax,
                         float* __restrict__ denom, int n) {
    int t = blockIdx.x * blockDim.x + threadIdx.x;
    if (t < n * 128) agg[t] = 0.f;
    if (t < n * 4) {
        emax[t]  = -__builtin_inff();
        denom[t] = 0.f;
    }
}

__device__ __forceinline__ void atomicMaxF(float* addr, float val) {
    if (val >= 0.f) atomicMax((int*)addr, __float_as_int(val));
    else            atomicMin((unsigned int*)addr, __float_as_uint(val));
}

__device__ __forceinline__ float leaky(float e) {
    return e > 0.f ? e : SLOPE * e;
}

// ---------------------------------------------------------------------------
// Kernel 4: per-edge logits + segment max over destinations
// Edges [0,E) come from edge_index; [E, E+n) are self-loops.
// ---------------------------------------------------------------------------
__global__ void gat_edge_max(const int* __restrict__ ei,
                             const float* __restrict__ a_src,
                             const float* __restrict__ a_dst,
                             float* __restrict__ emax, int E, int TE) {
    int t = blockIdx.x * blockDim.x + threadIdx.x;
    if (t >= TE) return;
    int s, d;
    if (t < E) { s = ei[t]; d = ei[E + t]; } else { s = d = t - E; }
    float4 as = *(const float4*)(a_src + 4 * (size_t)s);
    float4 ad = *(const float4*)(a_dst + 4 * (size_t)d);
    atomicMaxF(&emax[4 * (size_t)d + 0], leaky(as.x + ad.x));
    atomicMaxF(&emax[4 * (size_t)d + 1], leaky(as.y + ad.y));
    atomicMaxF(&emax[4 * (size_t)d + 2], leaky(as.z + ad.z));
    atomicMaxF(&emax[4 * (size_t)d + 3], leaky(as.w + ad.w));
}

// ---------------------------------------------------------------------------
// Kernel 5: e_exp per edge/head + segment-sum of denominators
// ---------------------------------------------------------------------------
__global__ void gat_edge_exp(const int* __restrict__ ei,
                             const float* __restrict__ a_src,
                             const float* __restrict__ a_dst,
                             const float* __restrict__ emax,
                             float* __restrict__ denom,
                             float* __restrict__ eexp, int E, int TE) {
    int t = blockIdx.x * blockDim.x + threadIdx.x;
    if (t >= TE) return;
    int s, d;
    if (t < E) { s = ei[t]; d = ei[E + t]; } else { s = d = t - E; }
    float4 as = *(const float4*)(a_src + 4 * (size_t)s);
    float4 ad = *(const float4*)(a_dst + 4 * (size_t)d);
    float4 em = *(const float4*)(emax + 4 * (size_t)d);
    float4 ex;
    ex.x = expf(leaky(as.x + ad.x) - em.x);
    ex.y = expf(leaky(as.y + ad.y) - em.y);
    ex.z = expf(leaky(as.z + ad.z) - em.z);
    ex.w = expf(leaky(as.w + ad.w) - em.w);
    *(float4*)(eexp + 4 * (size_t)t) = ex;
    unsafeAtomicAdd(&denom[4 * (size_t)d + 0], ex.x);
    unsafeAtomicAdd(&denom[4 * (size_t)d + 1], ex.y);
    unsafeAtomicAdd(&denom[4 * (size_t)d + 2], ex.z);
    unsafeAtomicAdd(&denom[4 * (size_t)d + 3], ex.w);
}

// ---------------------------------------------------------------------------
// Kernel 6: aggregation  agg[d,h,c] += alpha * xp[s,h,c]
// One wave per edge; lane owns float4 slice [4*lane, 4*lane+4); head = lane/8.
// ---------------------------------------------------------------------------
__global__ void gat_aggregate(const int* __restrict__ ei,
                              const float* __restrict__ eexp,
                              const float* __restrict__ denom,
                              const float* __restrict__ xp,
                              float* __restrict__ agg, int E, int TE) {
    int t = blockIdx.x * blockDim.x + threadIdx.x;
    int edge = t >> 5;
    int lane = t & 31;
    if (edge >= TE) return;
    int s, d;
    if (edge < E) { s = ei[edge]; d = ei[E + edge]; } else { s = d = edge - E; }
    int h = lane >> 3;
    float alpha = eexp[4 * (size_t)edge + h] / denom[4 * (size_t)d + h];
    float4 v = *(const float4*)(xp + (size_t)s * 128 + lane * 4);
    float* o = agg + (size_t)d * 128 + lane * 4;
    unsafeAtomicAdd(o + 0, alpha * v.x);
    unsafeAtomicAdd(o + 1, alpha * v.y);
    unsafeAtomicAdd(o + 2, alpha * v.z);
    unsafeAtomicAdd(o + 3, alpha * v.w);
}

// ---------------------------------------------------------------------------
// Kernel 7: head mean + bias + relu -> d_out[N,32]
// ---------------------------------------------------------------------------
__global__ void gat_finalize(const float* __restrict__ agg,
                             const float* __restrict__ bias,
                             float* __restrict__ out, int n) {
    int t = blockIdx.x * blockDim.x + threadIdx.x;
    if (t >= n * 32) return;
    int node = t >> 5, c = t & 31;
    const float* a = agg + (size_t)node * 128 + c;
    float v = (a[0] + a[32] + a[64] + a[96]) * 0.25f + bias[c];
    out[t] = v > 0.f ? v : 0.f;
}

extern "C" void kernel_launch(void* const* d_in, const int* in_sizes, int n_in,
                              void* d_out, int out_size, void* d_ws, size_t ws_size,
                              hipStream_t stream) {
    const float* x       = (const float*)d_in[0];
    const int*   ei      = (const int*)d_in[1];
    const float* W       = (const float*)d_in[2];
    const float* att_src = (const float*)d_in[3];
    const float* att_dst = (const float*)d_in[4];
    const float* bias    = (const float*)d_in[5];
    float* out = (float*)d_out;

    const int n  = in_sizes[0] / 128;     // nodes
    const int E  = in_sizes[1] / 2;       // directed edges
    const int TE = E + n;                 // + self loops

    // workspace layout (fp32), all 256B aligned
    char* p = (char*)d_ws;
    auto take = [&](size_t bytes) {
        char* r = p;
        p += (bytes + 255) & ~(size_t)255;
        return r;
    };
    float* xp    = (float*)take((size_t)n * 128 * 4);
    float* a_src = (float*)take((size_t)n * 4 * 4);
    float* a_dst = (float*)take((size_t)n * 4 * 4);
    float* emax  = (float*)take((size_t)n * 4 * 4);
    float* denom = (float*)take((size_t)n * 4 * 4);
    float* eexp  = (float*)take((size_t)TE * 4 * 4);
    float* agg   = (float*)take((size_t)n * 128 * 4);

    const int B = 256;
    // 1) projection GEMM (WMMA)
    gat_gemm_wmma<<<(n + 15) / 16, B, 0, stream>>>(x, W, xp, n);
    // 2) logits
    gat_logits<<<(n * 4 + B - 1) / B, B, 0, stream>>>(xp, att_src, att_dst,
                                                      a_src, a_dst, n);
    // 3) init accumulators
    gat_init<<<((size_t)n * 128 + B - 1) / B, B, 0, stream>>>(agg, emax, denom, n);
    // 4) segment max
    gat_edge_max<<<(TE + B - 1) / B, B, 0, stream>>>(ei, a_src, a_dst, emax, E, TE);
    // 5) exp + denominators
    gat_edge_exp<<<(TE + B - 1) / B, B, 0, stream>>>(ei, a_src, a_dst, emax,
                                                     denom, eexp, E, TE);
    // 6) weighted scatter-add
    gat_aggregate<<<((size_t)TE * 32 + B - 1) / B, B, 0, stream>>>(ei, eexp, denom,
                                                                   xp, agg, E, TE);
    // 7) head mean + bias + relu
    gat_finalize<<<(n * 32 + B - 1) / B, B, 0, stream>>>(agg, bias, out, n);
}